// ATLoss_84181359002214
// MI455X (gfx1250) — compile-verified
//
#include <hip/hip_runtime.h>
#include <math.h>

#define CLS 97           // relation classes
#define ROWS 8           // rows per span
#define SEG_DW (ROWS*CLS)   // 776 dwords per segment (3104 B, 16B-aligned stride)
#define SLOT_DW 800         // padded LDS slot (tail async op scribbles dwords 776..799)
#define WPB 8               // waves per block
#define BLOCK (WPB*32)
#define NBLK 512
#define TOTAL_WAVES (NBLK*WPB)   // 4096

// ---- CDNA5 async global->LDS copy path (ASYNCcnt) ------------------------

__device__ __forceinline__ unsigned lds_off(const void* p) {
  // flat shared pointers carry the LDS byte offset in the low 32 bits
  return (unsigned)(size_t)p;
}

__device__ __forceinline__ void async_b128(unsigned lds_b, unsigned off_b, const float* base) {
  asm volatile("global_load_async_to_lds_b128 %0, %1, %2"
               :: "v"(lds_b), "v"(off_b), "s"(base) : "memory");
}
__device__ __forceinline__ void async_b32(unsigned lds_b, unsigned off_b, const float* base) {
  asm volatile("global_load_async_to_lds_b32 %0, %1, %2"
               :: "v"(lds_b), "v"(off_b), "s"(base) : "memory");
}
__device__ __forceinline__ void wait_async_le7() {
  asm volatile("s_wait_asynccnt 0x7" ::: "memory");
}
__device__ __forceinline__ void wait_async_0() {
  asm volatile("s_wait_asynccnt 0x0" ::: "memory");
}
__device__ __forceinline__ void wait_ds0() {
  asm volatile("s_wait_dscnt 0x0" ::: "memory");
}

// ---- wave32 butterfly reductions -----------------------------------------

__device__ __forceinline__ float wave_max(float v) {
#pragma unroll
  for (int o = 16; o > 0; o >>= 1) v = fmaxf(v, __shfl_xor(v, o, 32));
  return v;
}
__device__ __forceinline__ float wave_sum(float v) {
#pragma unroll
  for (int o = 16; o > 0; o >>= 1) v += __shfl_xor(v, o, 32);
  return v;
}

// Stage one segment's 776 dwords into an LDS slot: 6x async B128 + 1x async B32 tail.
__device__ __forceinline__ void stage_segment(const float* __restrict__ logits,
                                              const int* __restrict__ pos,
                                              int seg, float* slot, int lane) {
  int st = __builtin_amdgcn_readfirstlane(pos[2 * seg]);
  const float* gbase = logits + (size_t)st * CLS;   // 16B-aligned (st*97*4 = st*388? no: st is 8j -> 8j*97*4 = j*3104, 16B aligned)
  unsigned lb = lds_off(slot);
  unsigned lo = (unsigned)lane * 16u;
#pragma unroll
  for (int k = 0; k < 6; ++k)
    async_b128(lb + (unsigned)k * 512u + lo, (unsigned)k * 512u + lo, gbase);
  // tail: dwords 768..775 (lanes 0..7); other lanes re-read dword 0 into unused LDS pad
  unsigned goff = (lane < 8) ? (3072u + (unsigned)lane * 4u) : 0u;
  async_b32(lb + 3072u + (unsigned)lane * 4u, goff, gbase);
}

__global__ __launch_bounds__(BLOCK) void atloss_seg_kernel(
    const float* __restrict__ logits, const float* __restrict__ labels,
    const int* __restrict__ pos, float* __restrict__ partial, int ep) {
  __shared__ __align__(16) float lds[WPB * 2 * SLOT_DW];   // 50 KB: double buffer per wave
  const int lane = threadIdx.x & 31;
  const int wid  = threadIdx.x >> 5;
  const int gw   = blockIdx.x * WPB + wid;

  float* slot0 = &lds[(wid * 2 + 0) * SLOT_DW];
  float* slot1 = &lds[(wid * 2 + 1) * SLOT_DW];

  int nseg = (gw < ep) ? ((ep - 1 - gw) / TOTAL_WAVES + 1) : 0;

  if (nseg > 0) stage_segment(logits, pos, gw, slot0, lane);
  if (nseg > 1) stage_segment(logits, pos, gw + TOTAL_WAVES, slot1, lane);

  const float NEG = -__builtin_inff();
  float acc = 0.0f;

  for (int i = 0; i < nseg; ++i) {
    const int s = gw + i * TOTAL_WAVES;
    if (i + 1 < nseg) wait_async_le7();   // prev 7-op batch done (in-order completion)
    else              wait_async_0();

    const float* cur = (i & 1) ? slot1 : slot0;
    int st = pos[2 * s], en = pos[2 * s + 1];
    int nr = en - st; if (nr > ROWS) nr = ROWS;
    const float th = logits[(size_t)s * CLS];   // logits[s, 0] (threshold override)

    // column-wise max over the span: lane owns cols {l, l+32, l+64}; col 96 broadcast
    float m0 = NEG, m1 = NEG, m2 = NEG, m96 = NEG;
    for (int r = 0; r < nr; ++r) {
      const float* row = cur + r * CLS;
      m0  = fmaxf(m0,  row[lane]);
      m1  = fmaxf(m1,  row[lane + 32]);
      m2  = fmaxf(m2,  row[lane + 64]);
      m96 = fmaxf(m96, row[96]);
    }
    const bool lane0 = (lane == 0);
    if (lane0) m0 = th;                         // e_logits[:,0] = logits[:ep,0]

    const float* lrow = labels + (size_t)s * CLS;
    float l0  = lane0 ? 0.0f : lrow[lane];      // labels[:,0] forced to 0
    float l1  = lrow[lane + 32];
    float l2  = lrow[lane + 64];
    float l96 = lrow[96];

    // ---- loss1: log-softmax over {class 0} U {positive classes} ----
    float a0 = (lane0 || l0 != 0.0f) ? m0 : NEG;
    float a1 = (l1 != 0.0f) ? m1 : NEG;
    float a2 = (l2 != 0.0f) ? m2 : NEG;
    float a3 = (lane0 && l96 != 0.0f) ? m96 : NEG;
    float P  = wave_max(fmaxf(fmaxf(a0, a1), fmaxf(a2, a3)));
    float s1 = expf(a0 - P) + expf(a1 - P) + expf(a2 - P) + expf(a3 - P); // exp(-inf)=0
    float Z1 = P + logf(wave_sum(s1));
    float pn = l0 + l1 + l2 + (lane0 ? l96 : 0.0f);
    float pm = l0 * m0 + l1 * m1 + l2 * m2 + (lane0 ? l96 * m96 : 0.0f);
    float loss1 = wave_sum(pn) * Z1 - wave_sum(pm);

    // ---- loss2: log-softmax over {class 0} U {negative classes}, pick class 0 ----
    float b0 = (l0 == 0.0f) ? m0 : NEG;         // class 0 included via lane0 (l0==0)
    float b1 = (l1 == 0.0f) ? m1 : NEG;
    float b2 = (l2 == 0.0f) ? m2 : NEG;
    float b3 = (lane0 && l96 == 0.0f) ? m96 : NEG;
    float Q  = wave_max(fmaxf(fmaxf(b0, b1), fmaxf(b2, b3)));
    float s2 = expf(b0 - Q) + expf(b1 - Q) + expf(b2 - Q) + expf(b3 - Q);
    float Z2 = Q + logf(wave_sum(s2));
    float loss2 = Z2 - th;

    acc += loss1 + loss2;

    if (i + 2 < nseg) {
      wait_ds0();   // LDS reads of this buffer fully drained before async overwrite
      stage_segment(logits, pos, gw + (i + 2) * TOTAL_WAVES, (i & 1) ? slot1 : slot0, lane);
    }
  }

  if (lane == 0) partial[gw] = acc;   // every wave writes (0 if no segments)
}

__global__ __launch_bounds__(256) void atloss_reduce_kernel(
    const float* __restrict__ partial, float* __restrict__ out, int n, float inv) {
  __shared__ float sm[256];
  float s = 0.0f;
  for (int i = threadIdx.x; i < n; i += 256) s += partial[i];
  sm[threadIdx.x] = s;
  __syncthreads();
#pragma unroll
  for (int o = 128; o > 0; o >>= 1) {
    if ((int)threadIdx.x < o) sm[threadIdx.x] += sm[threadIdx.x + o];
    __syncthreads();
  }
  if (threadIdx.x == 0) out[0] = sm[0] * inv;
}

extern "C" void kernel_launch(void* const* d_in, const int* in_sizes, int n_in,
                              void* d_out, int out_size, void* d_ws, size_t ws_size,
                              hipStream_t stream) {
  (void)n_in; (void)out_size; (void)ws_size;
  const float* logits = (const float*)d_in[0];
  const float* labels = (const float*)d_in[1];
  const int*   pos    = (const int*)d_in[2];
  const int ep = in_sizes[1] / CLS;          // labels is [EP, 97]
  float* partial = (float*)d_ws;             // TOTAL_WAVES floats = 16 KB

  atloss_seg_kernel<<<NBLK, BLOCK, 0, stream>>>(logits, labels, pos, partial, ep);
  atloss_reduce_kernel<<<1, 256, 0, stream>>>(partial, (float*)d_out, TOTAL_WAVES,
                                              1.0f / (float)ep);
}